// _SDN_Loss_51651276701971
// MI455X (gfx1250) — compile-verified
//
#include <hip/hip_runtime.h>

// SDN loss: bandwidth-bound streaming reduction.
// 224 MB read once @ 23.3 TB/s => ~9.6us floor. CDNA5 paths used:
//  - global_load_async_to_lds_b128 (ASYNCcnt) with NT hint + s_wait_asynccnt
//  - v_wmma_f32_16x16x4_f32 (A=ones) to finish the cross-thread reduction

#define TLEN  2048
#define TVEC  (TLEN / 4)       // 512 float4 per row
#define BLK   256
#define NARR  7
#define EPSV  1e-8f

typedef __attribute__((ext_vector_type(2))) float v2f;
typedef __attribute__((ext_vector_type(8))) float v8f;

__global__ __launch_bounds__(BLK) void sdn_row_kernel(
    const float* __restrict__ pred_start,
    const float* __restrict__ pred_end,
    const float* __restrict__ pred_mid,
    const float* __restrict__ pred_attw,
    const float* __restrict__ gt_start,
    const float* __restrict__ gt_end,
    const float* __restrict__ localiz,
    const int*   __restrict__ lens,
    float4* __restrict__ row_out)
{
    __shared__ float smem[NARR * TLEN];   // 56 KB staging tile
    __shared__ float red[6][BLK];         // 6 KB reduction scratch

    const int b   = blockIdx.x;
    const int tid = threadIdx.x;
    const long long rowoff = (long long)b * TLEN;

    const float* bases[NARR] = {
        pred_start + rowoff, pred_end + rowoff, pred_mid + rowoff,
        pred_attw + rowoff, gt_start + rowoff, gt_end + rowoff,
        localiz + rowoff };

    // LDS byte offset of the staging tile (low 32 bits of the generic addr).
    const unsigned lds0 = (unsigned)(unsigned long long)(&smem[0]);

    // Issue all 14 async B128 stream loads (coalesced, non-temporal),
    // zero VGPR payload pressure, one ASYNCcnt wait.
    #pragma unroll
    for (int a = 0; a < NARR; ++a) {
        #pragma unroll
        for (int j = 0; j < 2; ++j) {
            const int idx = tid * 4 + j * (BLK * 4);          // float index in row
            const unsigned ldsoff = lds0 + (unsigned)((a * TLEN + idx) * 4);
            const float* gaddr = bases[a] + idx;
            asm volatile("global_load_async_to_lds_b128 %0, %1, off th:TH_LOAD_NT"
                         :: "v"(ldsoff), "v"(gaddr) : "memory");
        }
    }
    asm volatile("s_wait_asynccnt 0x0" ::: "memory");
    __syncthreads();

    const int len = lens[b];

    // Per-row partial sums:
    //  s1 = sum gt_s * log(p_s/gt_s) * m          (KL start, negate later)
    //  s2 = sum gt_e * log(p_e/gt_e) * m          (KL end)
    //  s3 = sum loc  * log(p_attw + eps)          (UNMASKED, negate later)
    //  s4 = sum loc                               (UNMASKED)
    //  s5 = sum loc  * log(p_mid) * m             (negate later)
    //  s6 = sum loc * m
    float s1 = 0.f, s2 = 0.f, s3 = 0.f, s4 = 0.f, s5 = 0.f, s6 = 0.f;
    const float4* sp = (const float4*)smem;

    #pragma unroll
    for (int j = 0; j < 2; ++j) {
        const int v  = tid + j * BLK;   // float4 index within row
        const int t0 = v * 4;
        const float4 ps = sp[0 * TVEC + v];
        const float4 pe = sp[1 * TVEC + v];
        const float4 pm = sp[2 * TVEC + v];
        const float4 pa = sp[3 * TVEC + v];
        const float4 gs = sp[4 * TVEC + v];
        const float4 ge = sp[5 * TVEC + v];
        const float4 lo = sp[6 * TVEC + v];

        #define SDN_COMP(c, k) {                                              \
            const float m = ((t0 + (k)) < len) ? 1.0f : 0.0f;                 \
            s1 += gs.c * __logf(ps.c * __builtin_amdgcn_rcpf(gs.c)) * m;      \
            s2 += ge.c * __logf(pe.c * __builtin_amdgcn_rcpf(ge.c)) * m;      \
            s3 += lo.c * __logf(pa.c + EPSV);                                 \
            s4 += lo.c;                                                       \
            s5 += lo.c * __logf(pm.c) * m;                                    \
            s6 += lo.c * m; }
        SDN_COMP(x, 0) SDN_COMP(y, 1) SDN_COMP(z, 2) SDN_COMP(w, 3)
        #undef SDN_COMP
    }

    red[0][tid] = s1; red[1][tid] = s2; red[2][tid] = s3;
    red[3][tid] = s4; red[4][tid] = s5; red[5][tid] = s6;
    __syncthreads();

    // LDS tree: 256 -> 8 partials per quantity.
    #pragma unroll
    for (int off = BLK / 2; off >= 8; off >>= 1) {
        if (tid < off) {
            #pragma unroll
            for (int q = 0; q < 6; ++q) red[q][tid] += red[q][tid + off];
        }
        __syncthreads();
    }

    // Finish with 2 chained V_WMMA_F32_16X16X4_F32 ops.
    //   A = ones(16x4)  -> D[m][n] = sum_k B[k][n] + C[m][n]
    //   B[k][n] = red[n][k]  (quantity n in column n; 8 partials as K=0..7)
    // B layout (column-major, mirrors documented A layout): lane n (0-15)
    // holds {B[0][n], B[1][n]}, lane n+16 holds {B[2][n], B[3][n]}.
    // All waves execute this uniformly -> EXEC is all 1s (WMMA requirement).
    const int  n     = tid & 15;
    const int  hi    = (tid >> 4) & 1;     // lane half within the wave
    const bool qvld  = (n < 6);
    v2f a;  a.x = 1.0f; a.y = 1.0f;
    v2f b0, b1;
    b0.x = qvld ? red[n][hi * 2 + 0] : 0.0f;
    b0.y = qvld ? red[n][hi * 2 + 1] : 0.0f;
    b1.x = qvld ? red[n][4 + hi * 2 + 0] : 0.0f;
    b1.y = qvld ? red[n][4 + hi * 2 + 1] : 0.0f;
    v8f acc = {};
    acc = __builtin_amdgcn_wmma_f32_16x16x4_f32(false, a, false, b0,
                                                (short)0, acc, false, false);
    acc = __builtin_amdgcn_wmma_f32_16x16x4_f32(false, a, false, b1,
                                                (short)0, acc, false, false);
    // Every row of D is identical: acc[0] = total of quantity (tid & 15).
    const float tot = acc[0];
    const float t1 = __shfl(tot, 0, 32);   // s1 total
    const float t2 = __shfl(tot, 1, 32);   // s2 total
    const float t3 = __shfl(tot, 2, 32);   // s3 total
    const float t4 = __shfl(tot, 3, 32);   // s4 total
    const float t5 = __shfl(tot, 4, 32);   // s5 total
    const float t6 = __shfl(tot, 5, 32);   // s6 total

    if (tid == 0) {
        const float r_start = -t1;
        const float r_end   = -t2;
        const float r_attw  = (-t3) / (t4 + EPSV);
        const float r_mid   = 0.7f * (-t5) / (t6 + EPSV);
        row_out[b] = make_float4(r_start, r_end, r_attw, r_mid);
    }
}

__global__ __launch_bounds__(BLK) void sdn_final_kernel(
    const float4* __restrict__ rows, float* __restrict__ out, int nrows)
{
    __shared__ float4 red[BLK];
    const int tid = threadIdx.x;

    float4 acc = make_float4(0.f, 0.f, 0.f, 0.f);
    for (int i = tid; i < nrows; i += BLK) {     // fixed order -> deterministic
        const float4 v = rows[i];
        acc.x += v.x; acc.y += v.y; acc.z += v.z; acc.w += v.w;
    }
    red[tid] = acc;
    __syncthreads();

    #pragma unroll
    for (int off = BLK / 2; off > 0; off >>= 1) {
        if (tid < off) {
            const float4 o = red[tid + off];
            red[tid].x += o.x; red[tid].y += o.y;
            red[tid].z += o.z; red[tid].w += o.w;
        }
        __syncthreads();
    }

    if (tid == 0) {
        const float inv = 1.0f / (float)nrows;
        const float start_loss = red[0].x * inv;
        const float end_loss   = red[0].y * inv;
        const float attw_loss  = 0.5f * red[0].z * inv;
        const float mid_loss   = red[0].w * inv;
        out[0] = start_loss + end_loss + attw_loss + mid_loss;
        out[1] = start_loss;
        out[2] = end_loss;
        out[3] = attw_loss;
        out[4] = mid_loss;
    }
}

extern "C" void kernel_launch(void* const* d_in, const int* in_sizes, int n_in,
                              void* d_out, int out_size, void* d_ws, size_t ws_size,
                              hipStream_t stream) {
    const float* pred_start = (const float*)d_in[0];
    const float* pred_end   = (const float*)d_in[1];
    const float* pred_mid   = (const float*)d_in[2];
    const float* pred_attw  = (const float*)d_in[3];
    const float* gt_start   = (const float*)d_in[4];
    const float* gt_end     = (const float*)d_in[5];
    const float* localiz    = (const float*)d_in[6];
    const int*   lens       = (const int*)d_in[7];

    const int B = in_sizes[7];           // 4096 rows
    float4* rows = (float4*)d_ws;        // 64 KB of workspace

    sdn_row_kernel<<<B, BLK, 0, stream>>>(pred_start, pred_end, pred_mid,
                                          pred_attw, gt_start, gt_end,
                                          localiz, lens, rows);
    sdn_final_kernel<<<1, BLK, 0, stream>>>(rows, (float*)d_out, B);
}